// BertPerWordModel_35854386987575
// MI455X (gfx1250) — compile-verified
//
#include <hip/hip_runtime.h>

typedef __attribute__((ext_vector_type(2))) float v2f;
typedef __attribute__((ext_vector_type(8))) float v8f;

#define B_ 64
#define S_ 512
#define W_ 255
#define E_ 768

// Segment-mean as a tiny sparse GEMM on the WMMA pipe:
//   Out[b] (255x768) = Wmat[b] (255 x span, weights in {0, 0.5, 1.0}) * hs[b] (span x 768)
// One block = (batch b, tile of 16 words). 8 waves; each wave owns 6 column tiles of 16.
// Per column tile: burst-load 16 B values (immediate offsets off shared bases),
// 8x V_WMMA_F32_16X16X4_F32, then 8 stores off one base pointer with immediate offsets.
__global__ __launch_bounds__(256) void BertPerWordModel_segmean_wmma(
    const float* __restrict__ out_in,   // (B, S, E) f32
    const int*   __restrict__ mappings, // (B, W) int32 in {1,2}
    float*       __restrict__ dst)      // (B, W, E) f32
{
  const int wt   = blockIdx.x;          // word tile 0..15 (16 words each)
  const int b    = blockIdx.y;          // batch
  const int tid  = (int)threadIdx.x;
  const int lane = tid & 31;
  const int wave = tid >> 5;
  const int w0   = wt * 16;

  const int* __restrict__ mrow = mappings + b * W_;

  // ---- token base for this tile: sum of mappings[b][0..w0-1] (wave32 reduce) ----
  int acc = 0;
  for (int j = lane; j < w0; j += 32) acc += mrow[j];
  #pragma unroll
  for (int off = 16; off > 0; off >>= 1)
    acc += __builtin_amdgcn_ds_bpermute(((lane ^ off) << 2), acc);
  // all lanes now hold the tile's global token base

  // ---- per-word count + exclusive scan inside the 16-word tile ----
  const int r = lane & 15;              // word row / E-column within tile
  const int h = lane >> 4;              // half-wave id
  const int wIdx = w0 + r;
  const int mv = (wIdx < W_) ? mrow[wIdx] : 0;   // word 255 padded with 0
  int s = mv;                                    // inclusive scan over 16 lanes per half
  #pragma unroll
  for (int off = 1; off < 16; off <<= 1) {
    int t = __builtin_amdgcn_ds_bpermute(((lane - off) << 2), s);
    s += (r >= off) ? t : 0;
  }
  const int excl = s - mv;                       // word start, relative to tile token base
  const int span = __builtin_amdgcn_ds_bpermute((((lane & 16) | 15) << 2), s); // tokens in tile
  const float inv = (mv > 0) ? (1.0f / (float)mv) : 0.0f;  // exact: 1.0 or 0.5

  // ---- A fragments (16x4 f32 per chunk): lane holds M = lane%16, K = j + 2*(lane/16) ----
  v2f afrag[8];
  #pragma unroll
  for (int c = 0; c < 8; ++c) {
    #pragma unroll
    for (int j = 0; j < 2; ++j) {
      const int k = 4 * c + 2 * h + j;
      afrag[c][j] = (k >= excl && k < excl + mv) ? inv : 0.0f;
    }
  }

  const int e0base = wave * 96;        // 6 column tiles of 16 per wave

  // hs[b, t, :] == output[b, t+1, :]
  const float* __restrict__ src = out_in + (long)b * S_ * E_ + (long)(acc + 1) * E_
                                         + e0base + r;

  // ---- clamped per-slot row base pointers; tile offsets become immediates (it*64B) ----
  // slot t = 2*c + j holds token k = 4*c + 2*h + j for this lane's half.
  const float* rp[16];
  #pragma unroll
  for (int c = 0; c < 8; ++c) {
    #pragma unroll
    for (int j = 0; j < 2; ++j) {
      const int k  = 4 * c + 2 * h + j;
      const int kc = (k < span) ? k : (span - 1);   // in-bounds clamp; weight is 0 there
      rp[2 * c + j] = src + (long)kc * E_;
    }
  }

  // ---- single store base; all 48 store offsets are compile-time immediates ----
  // lane writes rows m = j + 8*h, col e0base + it*16 + r
  float* __restrict__ dbase = dst + (long)b * W_ * E_ + (long)(w0 + 8 * h) * E_
                                  + e0base + r;
  // only row m==15 (j==7, h==1) can be out of range (word index 255 in the last tile)
  const bool store7 = (w0 + 8 * h + 7) < W_;       // uniformly true for wt<15

  #pragma unroll
  for (int it = 0; it < 6; ++it) {
    // burst-load all 16 B values for this column tile (immediate offsets, one wait window)
    float bvl[16];
    #pragma unroll
    for (int t = 0; t < 16; ++t) bvl[t] = rp[t][it * 16];

    v8f cacc = {};
    #pragma unroll
    for (int c = 0; c < 8; ++c) {
      v2f bfrag;
      bfrag[0] = bvl[2 * c];
      bfrag[1] = bvl[2 * c + 1];
      cacc = __builtin_amdgcn_wmma_f32_16x16x4_f32(
          /*neg_a=*/false, afrag[c], /*neg_b=*/false, bfrag,
          /*c_mod=*/(short)0, cacc, /*reuse_a=*/false, /*reuse_b=*/false);
    }

    // C/D 16x16 f32: VGPR j holds row M = j + 8*(lane/16), col N = lane%16
    #pragma unroll
    for (int j = 0; j < 7; ++j)
      dbase[(long)j * E_ + it * 16] = cacc[j];     // always in range (m <= 14 < 255-240)
    if (store7)
      dbase[(long)7 * E_ + it * 16] = cacc[7];
  }
}

extern "C" void kernel_launch(void* const* d_in, const int* in_sizes, int n_in,
                              void* d_out, int out_size, void* d_ws, size_t ws_size,
                              hipStream_t stream) {
  (void)in_sizes; (void)n_in; (void)out_size; (void)d_ws; (void)ws_size;
  const float* out_in   = (const float*)d_in[0];   // (B,S,E) float32
  const int*   mappings = (const int*)d_in[1];     // (B,W) int32
  float*       dst      = (float*)d_out;           // (B,W,E) float32

  dim3 grid(16, B_);   // 16 word-tiles x 64 batches
  BertPerWordModel_segmean_wmma<<<grid, 256, 0, stream>>>(out_in, mappings, dst);
}